// MoM_6614249636016
// MI455X (gfx1250) — compile-verified
//
#include <hip/hip_runtime.h>

// ============================================================================
// MoM (Mixture-of-Memories) recurrence for MI455X (gfx1250, wave32, WMMA).
//
// Chunked linear attention: all projections and state updates are 16xKx{16,128}
// GEMMs on V_WMMA_F32_16X16X4_F32 (exact fp32; state accumulates 1024 rank-1
// updates so precision matters, and fp32 WMMA >> the ~40 GFLOP needed).
// Per-(batch,slot) 128x128 state lives in LDS. Gathers use CDNA5
// GLOBAL_LOAD_ASYNC_TO_LDS_B128 (ASYNCcnt) to move tiles straight into LDS.
// Deterministic: no atomics; lists built in t-order via ballot prefix sums.
//
// Workspace requirement: ~173 MB.
// ============================================================================

#define S_LEN   1024
#define BATCH   32
#define IN_DIM  512
#define DH      128
#define NMEM    16
#define NP1     17
#define RROWS   (S_LEN * BATCH)        // 32768
#define NLIST   (BATCH * NP1)          // 544

typedef float v2f __attribute__((ext_vector_type(2)));
typedef float v8f __attribute__((ext_vector_type(8)));

static __device__ inline v8f wmma4(v2f a, v2f b, v8f c) {
  // D(16x16,f32) = A(16x4,f32) * B(4x16,f32) + C
  return __builtin_amdgcn_wmma_f32_16x16x4_f32(
      /*neg_a=*/false, a, /*neg_b=*/false, b,
      /*c_mod=*/(short)0, c, /*reuse_a=*/false, /*reuse_b=*/false);
}

// --- CDNA5 async global->LDS copy (16B per lane), tracked by ASYNCcnt -------
typedef int v4i_g __attribute__((vector_size(16)));   // gcc-style int4

static __device__ inline void async_g2l_b128(void* lds_dst, const void* gsrc) {
#if __has_builtin(__builtin_amdgcn_global_load_async_to_lds_b128)
  typedef __attribute__((address_space(1))) v4i_g* gp_t;
  typedef __attribute__((address_space(3))) v4i_g* lp_t;
  __builtin_amdgcn_global_load_async_to_lds_b128((gp_t)gsrc, (lp_t)lds_dst, 0, 0);
#else
  const unsigned l = (unsigned)(unsigned long long)lds_dst;
  asm volatile("global_load_async_to_lds_b128 %0, %1, off"
               :: "v"(l), "v"(gsrc) : "memory");
#endif
}

static __device__ inline void wait_async0() {
#if __has_builtin(__builtin_amdgcn_s_wait_asynccnt)
  __builtin_amdgcn_s_wait_asynccnt(0);
#else
  asm volatile("s_wait_asynccnt 0x0" ::: "memory");
#endif
}

// ---------------------------------------------------------------------------
// Workspace layout (bytes)
// ---------------------------------------------------------------------------
static constexpr size_t OFF_SLOTS = 0;                                    // int2  [R]
static constexpr size_t OFF_GATES = OFF_SLOTS + (size_t)RROWS * 8;        // float2[R]
static constexpr size_t OFF_CNT   = OFF_GATES + (size_t)RROWS * 8;        // int   [544] (padded)
static constexpr size_t OFF_TLIST = OFF_CNT + 4096;                       // int   [544*1024]
static constexpr size_t OFF_GLIST = OFF_TLIST + (size_t)NLIST * S_LEN * 4;// float [544*1024]
static constexpr size_t OFF_Q     = OFF_GLIST + (size_t)NLIST * S_LEN * 4;// float [R*128]
static constexpr size_t OFF_KSEL  = OFF_Q    + (size_t)RROWS * DH * 4;    // float [R*3*128]
static constexpr size_t OFF_VSEL  = OFF_KSEL + (size_t)RROWS * 3 * DH * 4;
static constexpr size_t OFF_CSEL  = OFF_VSEL + (size_t)RROWS * 3 * DH * 4;

// ---------------------------------------------------------------------------
// Kernel A: gating.  scores = x @ W_g + b_g ; top-2 ; g1 = 1/(1+exp(z2-z1)).
// (softmax denominator cancels in the renormalized top-k gates)
// ---------------------------------------------------------------------------
__global__ void k_gate(const float* __restrict__ X, const float* __restrict__ Wg,
                       const float* __restrict__ bg, int2* __restrict__ slots,
                       float2* __restrict__ gates) {
  const int wave = threadIdx.x >> 5, lane = threadIdx.x & 31;
  const int r = blockIdx.x * 8 + wave;
  const float* xr = X + (size_t)r * IN_DIM;
  float acc[NMEM];
#pragma unroll
  for (int c = 0; c < NMEM; ++c) acc[c] = 0.f;
  for (int ii = 0; ii < IN_DIM / 32; ++ii) {
    const int i = ii * 32 + lane;
    const float xv = xr[i];
    const float* wrow = Wg + (size_t)i * NMEM;
#pragma unroll
    for (int c = 0; c < NMEM; ++c) acc[c] = fmaf(xv, wrow[c], acc[c]);
  }
#pragma unroll
  for (int c = 0; c < NMEM; ++c) {
    float v = acc[c];
#pragma unroll
    for (int off = 16; off >= 1; off >>= 1) v += __shfl_xor(v, off, 32);
    acc[c] = v + bg[c];
  }
  int s1 = 0;
#pragma unroll
  for (int c = 1; c < NMEM; ++c) if (acc[c] > acc[s1]) s1 = c;
  int s2 = (s1 == 0) ? 1 : 0;
#pragma unroll
  for (int c = 0; c < NMEM; ++c) if (c != s1 && acc[c] > acc[s2]) s2 = c;
  const float g1 = 1.f / (1.f + __expf(acc[s2] - acc[s1]));
  if (lane == 0) {
    slots[r] = make_int2(s1 + 1, s2 + 1);     // +1: shift past shared slot 0
    gates[r] = make_float2(g1, 1.f - g1);
  }
}

// ---------------------------------------------------------------------------
// Kernel A2: ordered selected-step lists per (b, slot); ballot prefix sums.
// Entry packing: t | (p << 16), p in {0 (shared), 1, 2}.
// ---------------------------------------------------------------------------
__global__ void k_lists(const int2* __restrict__ slots, const float2* __restrict__ gates,
                        int* __restrict__ tlist, float* __restrict__ glist,
                        int* __restrict__ cnt) {
  const int wave = threadIdx.x >> 5, lane = threadIdx.x & 31;
  const int L = blockIdx.x * 8 + wave;
  if (L >= NLIST) return;
  const int b = L / NP1, s = L % NP1;
  const unsigned ltmask = (1u << lane) - 1u;
  int count = 0;
  for (int base = 0; base < S_LEN; base += 32) {
    const int t = base + lane;
    const int2   sl = slots[t * BATCH + b];
    const float2 gg = gates[t * BATCH + b];
    int pred, p; float g;
    if (s == 0)            { pred = 1; p = 0; g = 1.f;  }
    else if (s == sl.x)    { pred = 1; p = 1; g = gg.x; }
    else if (s == sl.y)    { pred = 1; p = 2; g = gg.y; }
    else                   { pred = 0; p = 0; g = 0.f;  }
    const unsigned m = __builtin_amdgcn_ballot_w32(pred != 0);
    if (pred) {
      const int pos = count + __popc(m & ltmask);
      tlist[L * S_LEN + pos] = t | (p << 16);
      glist[L * S_LEN + pos] = g;
    }
    count += __popc(m);
  }
  if (lane == 0) cnt[L] = count;
}

// ---------------------------------------------------------------------------
// Kernel B: Q = X @ W_q + b_q via WMMA; X tile async-staged into LDS.
// ---------------------------------------------------------------------------
#define LDX 516  // 512 + 4 floats pad -> conflict-free A-fragment reads

__global__ void k_qproj(const float* __restrict__ X, const float* __restrict__ Wq,
                        const float* __restrict__ bq, float* __restrict__ Q) {
  __shared__ float ldsX[16 * LDX];
  const int tid = threadIdx.x;
  const int rbase = blockIdx.x * 16;
  {
    const int row = tid >> 4, seg = tid & 15;
    const float* src = X + (size_t)(rbase + row) * IN_DIM + seg * 32;
    float* dst = ldsX + row * LDX + seg * 32;
#pragma unroll
    for (int q = 0; q < 8; ++q) async_g2l_b128(dst + q * 4, src + q * 4);
  }
  wait_async0();
  __syncthreads();
  const int wave = tid >> 5, lane = tid & 31;
  const int colb = wave * 16;
  const int arow = lane & 15, hk = (lane >> 4) * 2, col = lane & 15;
  v8f c = {0.f, 0.f, 0.f, 0.f, 0.f, 0.f, 0.f, 0.f};
  for (int k0 = 0; k0 < IN_DIM; k0 += 4) {
    const int kk = k0 + hk;
    v2f a;  a.x = ldsX[arow * LDX + kk];            a.y = ldsX[arow * LDX + kk + 1];
    v2f bb; bb.x = Wq[(size_t)kk * DH + colb + col]; bb.y = Wq[(size_t)(kk + 1) * DH + colb + col];
    c = wmma4(a, bb, c);
  }
  const float bias = bq[colb + col];
#pragma unroll
  for (int jj = 0; jj < 8; ++jj) {
    const int row = jj + ((lane >> 4) << 3);
    Q[(size_t)(rbase + row) * DH + colb + col] = c[jj] + bias;
  }
}

// ---------------------------------------------------------------------------
// Kernel C: gathered K/V projection per (b,slot) list chunk; async X gather,
// WMMA against the slot's 512x128 weight blocks, scatter to Ksel/Vsel.
// ---------------------------------------------------------------------------
__global__ void k_kvproj(const float* __restrict__ X,
                         const float* __restrict__ Wk, const float* __restrict__ bk,
                         const float* __restrict__ Wv, const float* __restrict__ bv,
                         const int* __restrict__ tlist, const int* __restrict__ cnt,
                         float* __restrict__ Ksel, float* __restrict__ Vsel) {
  const int cidx = blockIdx.x, s = blockIdx.y, b = blockIdx.z;
  const int L = b * NP1 + s;
  const int m = cnt[L];
  const int base = cidx * 16;
  if (base >= m) return;                         // uniform exit
  __shared__ float ldsX[16 * LDX];
  __shared__ int entT[16];
  const int tid = threadIdx.x;
  if (tid < 16) entT[tid] = (base + tid < m) ? tlist[L * S_LEN + base + tid] : -1;
  __syncthreads();
  {
    const int row = tid >> 4, seg = tid & 15;
    const int e = entT[row];
    float* dst = ldsX + row * LDX + seg * 32;
    if (e >= 0) {
      const int t = e & 0xFFFF;
      const float* src = X + (size_t)(t * BATCH + b) * IN_DIM + seg * 32;
#pragma unroll
      for (int q = 0; q < 8; ++q) async_g2l_b128(dst + q * 4, src + q * 4);
    } else {
      const float4 z = {0.f, 0.f, 0.f, 0.f};
#pragma unroll
      for (int q = 0; q < 8; ++q) ((float4*)dst)[q] = z;
    }
  }
  wait_async0();
  __syncthreads();
  const int wave = tid >> 5, lane = tid & 31;
  const int colb = wave * 16;
  const int arow = lane & 15, hk = (lane >> 4) * 2, col = lane & 15;
  const float* WkS = Wk + s * DH;   // column block of (512 x 2176) weight
  const float* WvS = Wv + s * DH;
  v8f ck = {0.f, 0.f, 0.f, 0.f, 0.f, 0.f, 0.f, 0.f};
  v8f cv = {0.f, 0.f, 0.f, 0.f, 0.f, 0.f, 0.f, 0.f};
  for (int k0 = 0; k0 < IN_DIM; k0 += 4) {
    const int kk = k0 + hk;
    v2f a;  a.x = ldsX[arow * LDX + kk]; a.y = ldsX[arow * LDX + kk + 1];
    v2f b1; b1.x = WkS[(size_t)kk * (NP1 * DH) + colb + col];
    b1.y = WkS[(size_t)(kk + 1) * (NP1 * DH) + colb + col];
    ck = wmma4(a, b1, ck);
    v2f b2; b2.x = WvS[(size_t)kk * (NP1 * DH) + colb + col];
    b2.y = WvS[(size_t)(kk + 1) * (NP1 * DH) + colb + col];
    cv = wmma4(a, b2, cv);
  }
  const float biask = bk[s * DH + colb + col];
  const float biasv = bv[s * DH + colb + col];
#pragma unroll
  for (int jj = 0; jj < 8; ++jj) {
    const int row = jj + ((lane >> 4) << 3);
    const int e = entT[row];
    if (e >= 0) {
      const int t = e & 0xFFFF, p = (e >> 16) & 3;
      const size_t o = ((size_t)(t * BATCH + b) * 3 + p) * DH + colb + col;
      Ksel[o] = ck[jj] + biask;
      Vsel[o] = cv[jj] + biasv;
    }
  }
}

// ---------------------------------------------------------------------------
// Kernel D: persistent chunked recurrence.  One block per (slot, batch); the
// 128x128 fp32 state M lives in LDS.  Per 16-entry chunk:
//   S  = Qc Kc^T (causal, diag incl.) -> out = S Vc + Qc M -> M += Kc^T Vc
// Chunk Q/K/V rows are async-gathered straight into LDS.
// ---------------------------------------------------------------------------
#define LDC  132   // 128 + 4 pad: conflict-free Kc^T fragment reads
#define LDSS 20    // 16 + 4 pad for the score tile

__global__ void k_recur(const float* __restrict__ M0, const float* __restrict__ Q,
                        const float* __restrict__ Ksel, const float* __restrict__ Vsel,
                        const int* __restrict__ tlist, const float* __restrict__ glist,
                        const int* __restrict__ cnt, float* __restrict__ Csel) {
  extern __shared__ float lds[];
  float* Mlds = lds;                     // 128*128
  float* Qc   = lds + DH * DH;           // 16*LDC
  float* Kc   = Qc + 16 * LDC;
  float* Vc   = Kc + 16 * LDC;
  float* Ssc  = Vc + 16 * LDC;           // 16*LDSS
  int*   entT = (int*)(Ssc + 16 * LDSS); // 16
  float* gtF  = (float*)(entT + 16);     // 16

  const int s = blockIdx.x, b = blockIdx.y;
  const int L = b * NP1 + s;
  const int m = cnt[L];
  const int tid = threadIdx.x;
  {  // init state from M_0 (async global->LDS)
    const float* src = (const float*)M0;
    for (int q = tid; q < (DH * DH) / 4; q += 256)
      async_g2l_b128(Mlds + q * 4, src + q * 4);
  }
  wait_async0();
  __syncthreads();

  const int wave = tid >> 5, lane = tid & 31;
  const int arow = lane & 15, h = lane >> 4, hk = (lane >> 4) * 2, col = lane & 15;
  const int nch = (m + 15) >> 4;

  for (int cidx = 0; cidx < nch; ++cidx) {
    const int base = cidx * 16;
    if (tid < 16) {
      const bool valid = (base + tid) < m;
      entT[tid] = valid ? tlist[L * S_LEN + base + tid] : -1;
      gtF[tid]  = valid ? glist[L * S_LEN + base + tid] : 0.f;
    }
    __syncthreads();
    {  // gather Qc/Kc/Vc rows (async; zero-pad invalid)
      const int row = tid >> 4, seg = tid & 15;
      const int e = entT[row];
      float* dq = Qc + row * LDC + seg * 8;
      float* dk = Kc + row * LDC + seg * 8;
      float* dv = Vc + row * LDC + seg * 8;
      if (e >= 0) {
        const int t = e & 0xFFFF, p = (e >> 16) & 3;
        const size_t r = (size_t)(t * BATCH + b);
        const float* sq = Q + r * DH + seg * 8;
        const size_t o = (r * 3 + p) * DH + seg * 8;
        async_g2l_b128(dq,     sq);
        async_g2l_b128(dq + 4, sq + 4);
        async_g2l_b128(dk,     Ksel + o);
        async_g2l_b128(dk + 4, Ksel + o + 4);
        async_g2l_b128(dv,     Vsel + o);
        async_g2l_b128(dv + 4, Vsel + o + 4);
      } else {
        const float4 z = {0.f, 0.f, 0.f, 0.f};
        ((float4*)dq)[0] = z; ((float4*)dq)[1] = z;
        ((float4*)dk)[0] = z; ((float4*)dk)[1] = z;
        ((float4*)dv)[0] = z; ((float4*)dv)[1] = z;
      }
    }
    wait_async0();
    __syncthreads();

    // --- score tile S = Qc @ Kc^T (wave 0), causal mask col<=row, to LDS ---
    if (wave == 0) {
      v8f sc = {0.f, 0.f, 0.f, 0.f, 0.f, 0.f, 0.f, 0.f};
      for (int k0 = 0; k0 < DH; k0 += 4) {
        const int kk = k0 + hk;
        v2f a;  a.x = Qc[arow * LDC + kk];  a.y = Qc[arow * LDC + kk + 1];
        v2f bb; bb.x = Kc[col * LDC + kk];  bb.y = Kc[col * LDC + kk + 1];
        sc = wmma4(a, bb, sc);
      }
#pragma unroll
      for (int jj = 0; jj < 8; ++jj) {
        const int row = jj + (h << 3);
        Ssc[row * LDSS + col] = (col <= row) ? sc[jj] : 0.f;
      }
    }
    __syncthreads();

    // --- out = S@Vc + Qc@M, gate-scale, scatter to Csel -------------------
    const int colb = wave * 16;
    v8f acc = {0.f, 0.f, 0.f, 0.f, 0.f, 0.f, 0.f, 0.f};
#pragma unroll
    for (int k0 = 0; k0 < 16; k0 += 4) {                 // intra-chunk
      const int kk = k0 + hk;
      v2f a;  a.x = Ssc[arow * LDSS + kk]; a.y = Ssc[arow * LDSS + kk + 1];
      v2f bb; bb.x = Vc[kk * LDC + colb + col]; bb.y = Vc[(kk + 1) * LDC + colb + col];
      acc = wmma4(a, bb, acc);
    }
    for (int k0 = 0; k0 < DH; k0 += 4) {                 // inter: Qc @ M
      const int kk = k0 + hk;
      v2f a;  a.x = Qc[arow * LDC + kk]; a.y = Qc[arow * LDC + kk + 1];
      v2f bb; bb.x = Mlds[kk * DH + colb + col]; bb.y = Mlds[(kk + 1) * DH + colb + col];
      acc = wmma4(a, bb, acc);
    }
#pragma unroll
    for (int jj = 0; jj < 8; ++jj) {
      const int row = jj + (h << 3);
      const int e = entT[row];
      if (e >= 0) {
        const int t = e & 0xFFFF, p = (e >> 16) & 3;
        const size_t o = ((size_t)(t * BATCH + b) * 3 + p) * DH + colb + col;
        Csel[o] = acc[jj] * gtF[row];
      }
    }
    __syncthreads();   // all reads of M done before updating it

    // --- M += Kc^T @ Vc : wave handles d-block wave*16, all 8 e-blocks ----
    {
      const int dbase = wave * 16;
      for (int eb = 0; eb < 8; ++eb) {
        const int ebase = eb * 16;
        v8f cm;
#pragma unroll
        for (int jj = 0; jj < 8; ++jj)
          cm[jj] = Mlds[(dbase + jj + (h << 3)) * DH + ebase + col];
#pragma unroll
        for (int i0 = 0; i0 < 16; i0 += 4) {
          const int ii = i0 + hk;
          v2f a;  a.x = Kc[ii * LDC + dbase + arow];       // A[d][i] = Kc[i][d]
          a.y = Kc[(ii + 1) * LDC + dbase + arow];
          v2f bb; bb.x = Vc[ii * LDC + ebase + col];
          bb.y = Vc[(ii + 1) * LDC + ebase + col];
          cm = wmma4(a, bb, cm);
        }
#pragma unroll
        for (int jj = 0; jj < 8; ++jj)
          Mlds[(dbase + jj + (h << 3)) * DH + ebase + col] = cm[jj];
      }
    }
    __syncthreads();
  }
}

// ---------------------------------------------------------------------------
// Kernel E: out[t,b,:] = Csel[p=0] + Csel[p=1] + Csel[p=2]
// ---------------------------------------------------------------------------
__global__ void k_reduce(const float* __restrict__ Csel, float* __restrict__ out) {
  const size_t idx = (size_t)blockIdx.x * 256 + threadIdx.x;  // R*128 total
  const size_t r = idx >> 7, e = idx & 127;
  const size_t o = r * 3 * DH + e;
  out[idx] = Csel[o] + Csel[o + DH] + Csel[o + 2 * DH];
}

// ---------------------------------------------------------------------------
extern "C" void kernel_launch(void* const* d_in, const int* in_sizes, int n_in,
                              void* d_out, int out_size, void* d_ws, size_t ws_size,
                              hipStream_t stream) {
  (void)in_sizes; (void)n_in; (void)out_size; (void)ws_size;
  const float* X  = (const float*)d_in[0];
  const float* M0 = (const float*)d_in[1];
  const float* Wk = (const float*)d_in[2];
  const float* bk = (const float*)d_in[3];
  const float* Wv = (const float*)d_in[4];
  const float* bv = (const float*)d_in[5];
  const float* Wg = (const float*)d_in[6];
  const float* bg = (const float*)d_in[7];
  const float* Wq = (const float*)d_in[8];
  const float* bq = (const float*)d_in[9];
  // d_in[10] = k (top-k); reference fixes k=2, hardcoded in the pipeline.

  char* ws = (char*)d_ws;
  int2*   slots = (int2*)(ws + OFF_SLOTS);
  float2* gates = (float2*)(ws + OFF_GATES);
  int*    cnt   = (int*)(ws + OFF_CNT);
  int*    tlist = (int*)(ws + OFF_TLIST);
  float*  glist = (float*)(ws + OFF_GLIST);
  float*  Q     = (float*)(ws + OFF_Q);
  float*  Ksel  = (float*)(ws + OFF_KSEL);
  float*  Vsel  = (float*)(ws + OFF_VSEL);
  float*  Csel  = (float*)(ws + OFF_CSEL);
  float*  out   = (float*)d_out;

  k_gate  <<<RROWS / 8, 256, 0, stream>>>(X, Wg, bg, slots, gates);
  k_lists <<<NLIST / 8, 256, 0, stream>>>(slots, gates, tlist, glist, cnt);
  k_qproj <<<RROWS / 16, 256, 0, stream>>>(X, Wq, bq, Q);
  k_kvproj<<<dim3(S_LEN / 16, NP1, BATCH), 256, 0, stream>>>(
      X, Wk, bk, Wv, bv, tlist, cnt, Ksel, Vsel);
  constexpr size_t ldsD = (DH * DH + 3 * 16 * LDC + 16 * LDSS + 32) * sizeof(float);
  k_recur <<<dim3(NP1, BATCH), 256, ldsD, stream>>>(
      M0, Q, Ksel, Vsel, tlist, glist, cnt, Csel);
  k_reduce<<<(RROWS * DH) / 256, 256, 0, stream>>>(Csel, out);
}